// GraphSAGE_85813446574086
// MI455X (gfx1250) — compile-verified
//
#include <hip/hip_runtime.h>
#include <math.h>

typedef float v2f __attribute__((ext_vector_type(2)));
typedef float v4f __attribute__((ext_vector_type(4)));
typedef float v8f __attribute__((ext_vector_type(8)));

#define NNODES 100000
#define DIM 64
#define NS 20
#define LDT 68                 // padded LDS row stride in floats (68 % 64 == 4 -> conflict-free)
#define WAVES 4
#define NODES_PER_BLOCK (WAVES * 16)
#define EPSN 1e-12f

__global__ __launch_bounds__(128, 1) void sage_wmma_kernel(
    const int* __restrict__ nodes_u, const int* __restrict__ nodes_i,
    const int* __restrict__ nbr_u,   const int* __restrict__ nbr_i,
    const float* __restrict__ tab_u, const float* __restrict__ tab_i,
    const float* __restrict__ Wsu, const float* __restrict__ bsu,
    const float* __restrict__ Wnu, const float* __restrict__ bnu,
    const float* __restrict__ Wsi, const float* __restrict__ bsi,
    const float* __restrict__ Wni, const float* __restrict__ bni,
    float* __restrict__ out)
{
    __shared__ float lds_ws[DIM * LDT];
    __shared__ float lds_wn[DIM * LDT];
    __shared__ float lds_b[DIM];
    __shared__ float tiles[WAVES][16 * LDT];

    const int tid  = threadIdx.x;
    const int side = blockIdx.y;

    const int*   nodes = side ? nodes_i : nodes_u;
    const int*   nbrs  = side ? nbr_i   : nbr_u;
    const float* tab   = side ? tab_i   : tab_u;
    const float* Ws    = side ? Wsi : Wsu;
    const float* Wn    = side ? Wni : Wnu;
    const float* bsp   = side ? bsi : bsu;
    const float* bnp   = side ? bni : bnu;
    float* outp = out + (size_t)side * NNODES * DIM;

    // ---- stage weights + fused bias into LDS (once per block) ----
    for (int idx = tid; idx < DIM * DIM; idx += blockDim.x) {
        int r = idx >> 6, cc = idx & 63;
        lds_ws[r * LDT + cc] = Ws[idx];
        lds_wn[r * LDT + cc] = Wn[idx];
    }
    if (tid < DIM) lds_b[tid] = bsp[tid] + bnp[tid];
    __syncthreads();

    const int wave = tid >> 5;
    const int lane = tid & 31;
    const int tile_base = blockIdx.x * NODES_PER_BLOCK + wave * 16;
    float* mytile = tiles[wave];

    // gather roles: 2 lanes per row, each covers 32 contiguous floats
    const int grow  = lane >> 1;
    const int ghalf = lane & 1;
    int gnode_row = tile_base + grow;
    if (gnode_row >= NNODES) gnode_row = 0;   // clamp; garbage rows never stored

    // ---- neighbor gather + mean -> LDS tile ----
    v4f acc[8];
#pragma unroll
    for (int i = 0; i < 8; ++i) acc[i] = (v4f)0.0f;
    for (int s = 0; s < NS; ++s) {
        int nb = nbrs[gnode_row * NS + s];
        const v4f* src = (const v4f*)(tab + (size_t)nb * DIM + ghalf * 32);
#pragma unroll
        for (int i = 0; i < 8; ++i) acc[i] += src[i];
    }
    {
        const float inv_s = 1.0f / (float)NS;
        v4f* dst = (v4f*)(mytile + grow * LDT + ghalf * 32);
#pragma unroll
        for (int i = 0; i < 8; ++i) dst[i] = acc[i] * inv_s;
    }
    __syncthreads();

    // ---- C accumulators initialized with fused bias ----
    v8f c[4];
#pragma unroll
    for (int ct = 0; ct < 4; ++ct) {
        float bv = lds_b[ct * 16 + (lane & 15)];
#pragma unroll
        for (int i = 0; i < 8; ++i) c[ct][i] = bv;
    }

    // operand addressing (ISA 16x4 f32 A layout):
    // lanes 0-15 take K = k0,k0+1 ; lanes 16-31 take K = k0+2,k0+3
    const int koff = 2 * (lane >> 4);
    const int arow = lane & 15;

    // ---- neigh_mean @ Wn^T via v_wmma_f32_16x16x4_f32 ----
#pragma unroll
    for (int ct = 0; ct < 4; ++ct) {
        const float* ap = mytile + arow * LDT + koff;
        const float* bp = lds_wn + (ct * 16 + arow) * LDT + koff;  // B[k][j] = Wn[j][k]
#pragma unroll
        for (int k0 = 0; k0 < DIM; k0 += 4) {
            v2f a = *(const v2f*)(ap + k0);
            v2f b = *(const v2f*)(bp + k0);
            c[ct] = __builtin_amdgcn_wmma_f32_16x16x4_f32(
                false, a, false, b, (short)0, c[ct], false, false);
        }
    }
    __syncthreads();

    // ---- self gather -> same LDS tile ----
    {
        int nid = nodes[gnode_row];
        const v4f* src = (const v4f*)(tab + (size_t)nid * DIM + ghalf * 32);
        v4f* dst = (v4f*)(mytile + grow * LDT + ghalf * 32);
#pragma unroll
        for (int i = 0; i < 8; ++i) dst[i] = src[i];
    }
    __syncthreads();

    // ---- self @ Ws^T ----
#pragma unroll
    for (int ct = 0; ct < 4; ++ct) {
        const float* ap = mytile + arow * LDT + koff;
        const float* bp = lds_ws + (ct * 16 + arow) * LDT + koff;
#pragma unroll
        for (int k0 = 0; k0 < DIM; k0 += 4) {
            v2f a = *(const v2f*)(ap + k0);
            v2f b = *(const v2f*)(bp + k0);
            c[ct] = __builtin_amdgcn_wmma_f32_16x16x4_f32(
                false, a, false, b, (short)0, c[ct], false, false);
        }
    }

    // ---- ReLU in registers ----
#pragma unroll
    for (int ct = 0; ct < 4; ++ct)
#pragma unroll
        for (int i = 0; i < 8; ++i) c[ct][i] = fmaxf(c[ct][i], 0.0f);

    __syncthreads();

    // ---- spill h to LDS tile (C/D layout: VGPR i -> M=i (lanes 0-15), M=i+8 (lanes 16-31)) ----
    {
        const int mbase = (lane >> 4) << 3;
        const int col0  = lane & 15;
#pragma unroll
        for (int ct = 0; ct < 4; ++ct)
#pragma unroll
            for (int i = 0; i < 8; ++i)
                mytile[(i + mbase) * LDT + ct * 16 + col0] = c[ct][i];
    }
    __syncthreads();

    // ---- row L2 normalize + store ----
    {
        const int r = lane & 15;
        float ss = 0.0f;
#pragma unroll
        for (int k = 0; k < DIM; ++k) {
            float v = mytile[r * LDT + k];
            ss += v * v;
        }
        float scale = 1.0f / fmaxf(sqrtf(ss), EPSN);
        int row = tile_base + r;
        if (row < NNODES) {
            const int cb = (lane < 16) ? 0 : 32;
            v4f* gout = (v4f*)(outp + (size_t)row * DIM + cb);
            const v4f* src = (const v4f*)(mytile + r * LDT + cb);
#pragma unroll
            for (int i = 0; i < 8; ++i) gout[i] = src[i] * scale;
        }
    }
}

extern "C" void kernel_launch(void* const* d_in, const int* in_sizes, int n_in,
                              void* d_out, int out_size, void* d_ws, size_t ws_size,
                              hipStream_t stream) {
    const int*   user_nodes = (const int*)d_in[0];
    const int*   item_nodes = (const int*)d_in[1];
    const int*   user_nbr   = (const int*)d_in[2];
    const int*   item_nbr   = (const int*)d_in[3];
    const float* user_tab   = (const float*)d_in[4];
    const float* item_tab   = (const float*)d_in[5];
    const float* Wsu = (const float*)d_in[6];
    const float* bsu = (const float*)d_in[7];
    const float* Wnu = (const float*)d_in[8];
    const float* bnu = (const float*)d_in[9];
    const float* Wsi = (const float*)d_in[10];
    const float* bsi = (const float*)d_in[11];
    const float* Wni = (const float*)d_in[12];
    const float* bni = (const float*)d_in[13];
    float* out = (float*)d_out;

    dim3 grid((NNODES + NODES_PER_BLOCK - 1) / NODES_PER_BLOCK, 2, 1);
    sage_wmma_kernel<<<grid, 128, 0, stream>>>(
        user_nodes, item_nodes, user_nbr, item_nbr,
        user_tab, item_tab,
        Wsu, bsu, Wnu, bnu, Wsi, bsi, Wni, bni,
        out);
}